// HashTopK_2791728742936
// MI455X (gfx1250) — compile-verified
//
#include <hip/hip_runtime.h>
#include <hip/hip_bf16.h>

// HashTopK router for MI455X (gfx1250).
// Memory-bound gather kernel: no matrix ops exist in this workload, so the
// optimization targets are the dependent-load chain (ids -> tid2eid -> logits
// gather), store coalescing, and keeping VALU work tiny so waves return to the
// memory queue quickly.
//
//   reads  ~180 MB (scattered logit gather dominates; logits array is 256 MB,
//           exceeding the 192 MB L2, so rows stream from HBM)
//   writes ~19 MB
//   => ~200 MB / 23.3 TB/s ~ 9 us floor.
//
// VALU path uses hardware transcendentals (v_exp_f32 / v_log_f32 / v_sqrt_f32
// / v_rcp_f32) instead of libm expansions.

#define NUM_EXPERTS 256
#define K_ROUTED    8
#define TOPK        9
#define BLOCK       256

// Device pass (__AMDGCN__): amdgcn hardware builtins.
// Host pass: generic clang builtins (typecheck only; never executed).
#if defined(__AMDGCN__)
#define FAST_EXP2(x) __builtin_amdgcn_exp2f(x)
#define FAST_LOG2(x) __builtin_amdgcn_logf(x)
#define FAST_SQRT(x) __builtin_amdgcn_sqrtf(x)
#define FAST_RCP(x)  __builtin_amdgcn_rcpf(x)
#else
#define FAST_EXP2(x) __builtin_exp2f(x)
#define FAST_LOG2(x) __builtin_log2f(x)
#define FAST_SQRT(x) __builtin_sqrtf(x)
#define FAST_RCP(x)  (1.0f / (x))
#endif

#define LOG2E_F 1.44269504088896340736f
#define LN2_F   0.69314718055994530942f

__global__ __launch_bounds__(BLOCK)
void hash_topk_kernel(const float* __restrict__ router_logits,
                      const int*   __restrict__ input_ids,
                      const int*   __restrict__ tid2eid,
                      float*       __restrict__ out_w,
                      int*         __restrict__ out_id,
                      int num_token)
{
    __shared__ float lds_w[BLOCK * TOPK];  // 9 KB
    __shared__ int   lds_i[BLOCK * TOPK];  // 9 KB (of 320 KB/WGP)

    const int lt = threadIdx.x;
    const int t  = blockIdx.x * BLOCK + lt;

    if (t < num_token) {
        const int tok = input_ids[t];
        const float* __restrict__ row = router_logits + (size_t)t * NUM_EXPERTS;

        // Prefetch the full 1 KB logits row (8 x 128B lines) so the row is
        // in-flight (into GL2) while we resolve the tid2eid gather. Emits
        // global_prefetch_b8 on gfx1250; shortens the 3-deep dependent chain.
        #pragma unroll
        for (int k = 0; k < 8; ++k)
            __builtin_prefetch(row + k * 32, 0, 3);

        // 32B-aligned 8-int row -> two global_load_b128
        const int4* __restrict__ erow = (const int4*)(tid2eid + (size_t)tok * K_ROUTED);
        const int4 e0 = erow[0];
        const int4 e1 = erow[1];
        int eid[K_ROUTED] = {e0.x, e0.y, e0.z, e0.w, e1.x, e1.y, e1.z, e1.w};

        // w_i = sqrt(softplus(logit)) for the 8 gathered experts only.
        // Stable softplus via hardware exp2/log2:
        //   softplus(x) = max(x,0) + ln2 * log2(1 + exp2(-|x| * log2e))
        float w[K_ROUTED];
        float sum = 0.0f;
        #pragma unroll
        for (int i = 0; i < K_ROUTED; ++i) {
            float x  = row[eid[i]];
            float e  = FAST_EXP2(-fabsf(x) * LOG2E_F);
            float sp = fmaxf(x, 0.0f) + LN2_F * FAST_LOG2(1.0f + e);
            w[i] = FAST_SQRT(sp);
            sum += w[i];
        }

        const float inv = FAST_RCP(sum);    // v_rcp_f32, ~1 ulp
        float s2 = 0.0f;
        #pragma unroll
        for (int i = 0; i < K_ROUTED; ++i) {
            w[i] *= inv;
            s2   += w[i];                   // faithful to reference: re-sum normalized
        }
        const float shared_w = s2 * (1.0f / 1.5f);

        // Stage the 9-wide rows in LDS; 9*lt+i mod 64 is conflict-free (gcd(9,64)=1)
        #pragma unroll
        for (int i = 0; i < K_ROUTED; ++i) {
            lds_w[lt * TOPK + i] = w[i];
            lds_i[lt * TOPK + i] = eid[i];
        }
        lds_w[lt * TOPK + K_ROUTED] = shared_w;
        lds_i[lt * TOPK + K_ROUTED] = NUM_EXPERTS;  // fused shared expert id = 256
    }

    __syncthreads();

    // Coalesced block-transposed writes: lane lt stores element j*256+lt of the
    // block's contiguous [BLOCK*TOPK] output span.
    const size_t base  = (size_t)blockIdx.x * (BLOCK * TOPK);
    const size_t limit = (size_t)num_token * TOPK;
    #pragma unroll
    for (int j = 0; j < TOPK; ++j) {
        const int    idx = j * BLOCK + lt;
        const size_t g   = base + idx;
        if (g < limit) {
            out_w[g]  = lds_w[idx];
            out_id[g] = lds_i[idx];
        }
    }
}

extern "C" void kernel_launch(void* const* d_in, const int* in_sizes, int n_in,
                              void* d_out, int out_size, void* d_ws, size_t ws_size,
                              hipStream_t stream)
{
    const float* router_logits = (const float*)d_in[0];   // [T, 256] f32
    const int*   input_ids     = (const int*)  d_in[1];   // [T] i32
    const int*   tid2eid       = (const int*)  d_in[2];   // [VOCAB, 8] i32

    const int num_token = in_sizes[1];

    // d_out = [topk_weights (T*9 f32) | topk_ids (T*9 i32, bit-stored)]
    float* out_w  = (float*)d_out;
    int*   out_id = (int*)d_out + (size_t)num_token * TOPK;

    const int grid = (num_token + BLOCK - 1) / BLOCK;
    hipLaunchKernelGGL(hash_topk_kernel, dim3(grid), dim3(BLOCK), 0, stream,
                       router_logits, input_ids, tid2eid, out_w, out_id, num_token);
}